// SkipGramNegativeSampling_4913442586663
// MI455X (gfx1250) — compile-verified
//
#include <hip/hip_runtime.h>

// ---------------------------------------------------------------------------
// Skip-gram negative sampling loss on MI455X (gfx1250).
// Memory-bound gather+dot: V_WMMA_F32_16X16X4_F32 with the diagonal trick
// (16 batch rows per wave; slot-j WMMA diag = the 16 per-row scores).
// Round 2: force global_load_b64 (AS1 casts) and group all 12 fragment loads
// per K-step ahead of the WMMA burst for latency hiding / partial waits.
// ---------------------------------------------------------------------------

typedef __attribute__((ext_vector_type(2))) float v2f;
typedef __attribute__((ext_vector_type(8))) float v8f;

#define BATCH   32768
#define DIM     512
#define KNEG    10
#define NSLOT   11          // 1 positive + 10 negatives
#define TILE    16          // batch rows per wave (WMMA M/N)
#define WPB     8           // waves per 256-thread block

// Explicit global-address-space load: guarantees global_load_b64 (LOADcnt only,
// no flat/DScnt coupling, no s_wait_xcnt pointer-update stalls).
typedef const v2f __attribute__((address_space(1)))* gv2f_p;
__device__ __forceinline__ v2f gload_v2(const float* p, int d) {
    return *(gv2f_p)(unsigned long long)(p + 4 * d);
}

// Select component r (0..7) of a v8f with constant-index selects (stays in VGPRs).
__device__ __forceinline__ float pick8(v8f c, int r) {
    float x = c[0];
    x = (r == 1) ? c[1] : x;
    x = (r == 2) ? c[2] : x;
    x = (r == 3) ? c[3] : x;
    x = (r == 4) ? c[4] : x;
    x = (r == 5) ? c[5] : x;
    x = (r == 6) ? c[6] : x;
    x = (r == 7) ? c[7] : x;
    return x;
}

// softplus(x) = log(1+exp(x)); inputs are clipped to [-10,10] so this is safe
// in f32 with the fast v_exp_f32 / v_log_f32 transcendentals.
__device__ __forceinline__ float softplus_fast(float x) {
    return __logf(1.0f + __expf(x));
}

__global__ void sgns_zero_out(float* out) { out[0] = 0.0f; }

__global__ __launch_bounds__(256) void sgns_wmma_kernel(
    const int*   __restrict__ center_ids,   // [B]
    const int*   __restrict__ context_ids,  // [B]
    const int*   __restrict__ neg_ids,      // [B,K]
    const float* __restrict__ center_emb,   // [V,D]
    const float* __restrict__ context_emb,  // [V,D]
    float*       __restrict__ out)          // [1]
{
    const unsigned lane = threadIdx.x & 31u;
    const unsigned wave = threadIdx.x >> 5;
    const unsigned tile = blockIdx.x * WPB + wave;

    // Fragment geometry for f32 16x16x4 A/B:
    //   lanes 0-15  hold (row = lane,    K = {k, k+1})
    //   lanes 16-31 hold (row = lane-16, K = {k+2, k+3})
    const unsigned rowInTile = lane & 15u;
    const unsigned row       = tile * TILE + rowInTile;   // this lane's batch row
    const unsigned c2        = (lane >> 4) * 2u;          // K sub-offset

    // A stream: center embedding rows.
    const float* ap = center_emb + (size_t)center_ids[row] * DIM + c2;

    // B streams: 11 context rows per batch row (column n == batch row n).
    const float* bp[NSLOT];
    bp[0] = context_emb + (size_t)context_ids[row] * DIM + c2;
#pragma unroll
    for (int j = 1; j < NSLOT; ++j)
        bp[j] = context_emb + (size_t)neg_ids[row * KNEG + (j - 1)] * DIM + c2;

    v8f acc[NSLOT];
#pragma unroll
    for (int j = 0; j < NSLOT; ++j) acc[j] = v8f{};

    // March K=0..511 in steps of 4. Issue ALL 12 fragment loads of the step
    // first (12 global_load_b64 in flight), then the 11-WMMA burst; the A
    // fragment is reused across all 11 slots.
#pragma unroll 2
    for (int d = 0; d < DIM / 4; ++d) {
        v2f a = gload_v2(ap, d);
        v2f b[NSLOT];
#pragma unroll
        for (int j = 0; j < NSLOT; ++j)
            b[j] = gload_v2(bp[j], d);
#pragma unroll
        for (int j = 0; j < NSLOT; ++j) {
            // (neg_a, A, neg_b, B, c_mod, C, reuse_a, reuse_b)
            acc[j] = __builtin_amdgcn_wmma_f32_16x16x4_f32(
                false, a, false, b[j], (short)0, acc[j], false, false);
        }
    }

    // Diagonal extraction: C[m][m] lives in
    //   lanes 0-7   -> m = lane,      component r = lane
    //   lanes 24-31 -> m = lane - 16, component r = lane - 24
    const bool isdiag = (lane < 8u) || (lane >= 24u);
    const int  r      = (lane < 8u) ? (int)lane : (int)lane - 24;

    float s0 = pick8(acc[0], r);
    s0 = fminf(fmaxf(s0, -10.0f), 10.0f);
    float loss = softplus_fast(-s0);            // -log_sigmoid(score)
#pragma unroll
    for (int j = 1; j < NSLOT; ++j) {
        float s = pick8(acc[j], r);
        s = fminf(fmaxf(s, -10.0f), 10.0f);
        loss += softplus_fast(s);               // -log_sigmoid(-neg_score)
    }
    loss = isdiag ? loss : 0.0f;

    // Wave32 reduction (16 live lanes -> lane 0).
#pragma unroll
    for (int off = 16; off >= 1; off >>= 1)
        loss += __shfl_xor(loss, off, 32);

    if (lane == 0)
        atomicAdd(out, loss * (1.0f / (float)BATCH));
}

extern "C" void kernel_launch(void* const* d_in, const int* in_sizes, int n_in,
                              void* d_out, int out_size, void* d_ws, size_t ws_size,
                              hipStream_t stream) {
    (void)in_sizes; (void)n_in; (void)out_size; (void)d_ws; (void)ws_size;

    const int*   center_ids  = (const int*)d_in[0];
    const int*   context_ids = (const int*)d_in[1];
    const int*   neg_ids     = (const int*)d_in[2];
    const float* center_emb  = (const float*)d_in[3];
    const float* context_emb = (const float*)d_in[4];
    float*       out         = (float*)d_out;

    sgns_zero_out<<<1, 1, 0, stream>>>(out);

    const int tiles  = BATCH / TILE;          // 2048 wave-tiles
    const int blocks = tiles / WPB;           // 256 blocks x 8 waves
    sgns_wmma_kernel<<<blocks, 256, 0, stream>>>(
        center_ids, context_ids, neg_ids, center_emb, context_emb, out);
}